// TextGNN_70463233459007
// MI455X (gfx1250) — compile-verified
//
#include <hip/hip_runtime.h>
#include <hip/hip_bf16.h>
#include <stdint.h>

// ---------------------------------------------------------------------------
// TextGNN (2-layer GCN + global mean pool) for gfx1250 (MI455X, wave32, WMMA)
// ---------------------------------------------------------------------------

typedef float v2f __attribute__((ext_vector_type(2)));
typedef float v4f __attribute__((ext_vector_type(4)));
typedef float v8f __attribute__((ext_vector_type(8)));

#define N_GRAPHS 256
#define OUT_DIM  64

// ---------------- degree / norm ----------------

__global__ void gnn_init_deg(float* __restrict__ deg, int n) {
    int i = blockIdx.x * blockDim.x + threadIdx.x;
    if (i < n) deg[i] = 1.0f;  // self-loop
}

__global__ void gnn_accum_deg(const long long* __restrict__ dst, float* __restrict__ deg, int E) {
    int e = blockIdx.x * blockDim.x + threadIdx.x;
    if (e < E) unsafeAtomicAdd(&deg[(int)dst[e]], 1.0f);
}

__global__ void gnn_rsqrt(float* __restrict__ deg, int n) {
    int i = blockIdx.x * blockDim.x + threadIdx.x;
    if (i < n) deg[i] = rsqrtf(deg[i]);
}

// ---------------- fp32 WMMA GEMM: C[M,N] = A[M,K] @ B[K,N] ------------------
// Block = 4 waves. Block stages a K x 32 column panel of B into LDS,
// pair-interleaved by K so each lane's WMMA B-operand pair {B[k][n],B[k+1][n]}
// is one aligned ds_load_b64. Each wave computes a 16x32 C tile (2 accs),
// reusing each A pair across two WMMAs. K is a compile-time constant (128),
// fully unrolled -> 64 v_wmma per body with immediate offsets.
//
// WMMA operand layouts (V_WMMA_F32_16X16X4_F32, wave32):
//   A 16x4 : lane l -> row (l&15); VGPR j -> K = kbase + 2*(l>>4) + j
//   B 4x16 : VGPR j, lane l -> K = kbase + 2*(l>>4) + j, col (l&15)
//   C/D    : VGPR r, lane l -> row r + 8*(l>>4), col (l&15)

template <int K>
__global__ void gnn_gemm_wmma(const float* __restrict__ A, const float* __restrict__ B,
                              float* __restrict__ C, int M, int N) {
    __shared__ float Bs[K * 32];  // Bs[(k>>1)*64 + c*2 + (k&1)], c = col - col0

    const int col0 = blockIdx.y * 32;
    const int tid  = threadIdx.y * 32 + threadIdx.x;

    // ---- stage B panel (K x 32) into LDS, pair-interleaved ----
#pragma unroll
    for (int idx = 0; idx < K * 32 / 128; ++idx) {
        const int e = tid + idx * 128;
        const int k = e >> 5;
        const int c = e & 31;
        Bs[(k >> 1) * 64 + c * 2 + (k & 1)] = B[(size_t)k * N + col0 + c];
    }
    __syncthreads();

    const int tileM = blockIdx.x * 4 + threadIdx.y;
    if (tileM * 16 >= M) return;  // uniform per wave

    const int lane = threadIdx.x;
    const int half = lane >> 4;   // 0 or 1
    const int lm   = lane & 15;

    const float* Arow = A + (size_t)(tileM * 16 + lm) * K + 2 * half;
    const float* Bp0  = Bs + half * 64 + lm * 2;         // col lm
    const float* Bp1  = Bs + half * 64 + (lm + 16) * 2;  // col lm+16

    v8f acc0 = {}, acc1 = {};
#pragma unroll
    for (int k = 0; k < K; k += 4) {
        const v2f a  = *(const v2f*)(Arow + k);            // global b64
        const v2f b0 = *(const v2f*)(Bp0 + (k >> 1) * 64); // ds b64
        const v2f b1 = *(const v2f*)(Bp1 + (k >> 1) * 64); // ds b64
        acc0 = __builtin_amdgcn_wmma_f32_16x16x4_f32(false, a, false, b0,
                                                     (short)0, acc0, false, false);
        acc1 = __builtin_amdgcn_wmma_f32_16x16x4_f32(false, a, false, b1,
                                                     (short)0, acc1, false, false);
    }

    float* Crow = C + (size_t)(tileM * 16 + 8 * half) * N + col0 + lm;
#pragma unroll
    for (int r = 0; r < 8; ++r) {
        Crow[(size_t)r * N]      = acc0[r];
        Crow[(size_t)r * N + 16] = acc1[r];
    }
}

// ---------------- edge scatter: acc[dst] += h[src] * dinv[src]*dinv[dst] ----

template <int F>
__global__ void gnn_scatter_edges(const float* __restrict__ h,
                                  const long long* __restrict__ src,
                                  const long long* __restrict__ dst,
                                  const float* __restrict__ dinv,
                                  float* __restrict__ acc, int E) {
    const int e = blockIdx.x * blockDim.y + threadIdx.y;  // one edge per wave
    if (e >= E) return;
    const int s = (int)src[e];
    const int d = (int)dst[e];
    const float norm = dinv[s] * dinv[d];
    const float* __restrict__ hs = h + (size_t)s * F;
    float* __restrict__ ad       = acc + (size_t)d * F;

    if constexpr (F == 128) {
        const v4f v = ((const v4f*)hs)[threadIdx.x];       // global b128
        const int f = threadIdx.x * 4;
        unsafeAtomicAdd(&ad[f + 0], v.x * norm);
        unsafeAtomicAdd(&ad[f + 1], v.y * norm);
        unsafeAtomicAdd(&ad[f + 2], v.z * norm);
        unsafeAtomicAdd(&ad[f + 3], v.w * norm);
    } else {
        const v2f v = ((const v2f*)hs)[threadIdx.x];       // global b64
        const int f = threadIdx.x * 2;
        unsafeAtomicAdd(&ad[f + 0], v.x * norm);
        unsafeAtomicAdd(&ad[f + 1], v.y * norm);
    }
}

// ---------------- epilogue: out = [relu](acc + h*dinv^2 + b) ----------------

template <int F, bool RELU>
__global__ void gnn_fuse_bias(const float* __restrict__ acc, const float* __restrict__ h,
                              const float* __restrict__ dinv, const float* __restrict__ b,
                              float* __restrict__ out, int n) {
    const int idx = blockIdx.x * blockDim.x + threadIdx.x;
    if (idx >= n * F) return;
    const int i = idx / F;
    const int f = idx - i * F;
    const float di = dinv[i];
    float v = acc[idx] + h[idx] * (di * di) + b[f];
    if (RELU) v = fmaxf(v, 0.0f);
    out[idx] = v;
}

// ---------------- global mean pool ----------------

__global__ void gnn_pool_accum(const float* __restrict__ h, const long long* __restrict__ batch,
                               float* __restrict__ out, float* __restrict__ cnt, int n) {
    const int i = blockIdx.x;       // one node per block
    const int f = threadIdx.x;      // 64 features
    if (i >= n) return;
    const int g = (int)batch[i];
    unsafeAtomicAdd(&out[g * OUT_DIM + f], h[(size_t)i * OUT_DIM + f]);
    if (f == 0) unsafeAtomicAdd(&cnt[g], 1.0f);
}

__global__ void gnn_pool_div(float* __restrict__ out, const float* __restrict__ cnt) {
    const int g = blockIdx.x;
    const int f = threadIdx.x;
    out[g * OUT_DIM + f] /= fmaxf(cnt[g], 1.0f);
}

// ---------------------------------------------------------------------------

extern "C" void kernel_launch(void* const* d_in, const int* in_sizes, int n_in,
                              void* d_out, int out_size, void* d_ws, size_t ws_size,
                              hipStream_t stream) {
    const float*     x     = (const float*)d_in[0];
    const long long* eidx  = (const long long*)d_in[1];
    const long long* batch = (const long long*)d_in[2];
    const float*     W1    = (const float*)d_in[3];
    const float*     b1    = (const float*)d_in[4];
    const float*     W2    = (const float*)d_in[5];
    const float*     b2    = (const float*)d_in[6];
    float*           out   = (float*)d_out;

    const int IN_DIM  = 128;
    const int HID_DIM = 128;
    const int N = in_sizes[0] / IN_DIM;   // 50000
    const int E = in_sizes[1] / 2;        // 800000
    const long long* srcp = eidx;
    const long long* dstp = eidx + E;

    // Workspace layout (floats), buffers reused across phases:
    //   dinv:  N                     (degree -> rsqrt)
    //   bufA:  N*128  (h1, then h2[N*64])
    //   bufB:  N*128  (acc1, then acc2[N*64])
    //   bufC:  N*128  (relu(h1'), then final layer-2 features[N*64])
    //   cnt:   N_GRAPHS
    float* ws   = (float*)d_ws;
    size_t oA   = ((size_t)N + 63) & ~(size_t)63;
    float* dinv = ws;
    float* bufA = ws + oA;
    float* bufB = bufA + (size_t)N * 128;
    float* bufC = bufB + (size_t)N * 128;
    float* cnt  = bufC + (size_t)N * 128;

    // ---- degree & symmetric norm ----
    gnn_init_deg<<<(N + 255) / 256, 256, 0, stream>>>(dinv, N);
    gnn_accum_deg<<<(E + 255) / 256, 256, 0, stream>>>(dstp, dinv, E);
    gnn_rsqrt<<<(N + 255) / 256, 256, 0, stream>>>(dinv, N);

    const int tilesM  = N / 16;           // 3125 (N divisible by 16)
    const int gridM   = (tilesM + 3) / 4; // 4 waves (16-row tiles) per block

    // ---- layer 1: h1 = x @ W1  (WMMA fp32, 16x32 wave tiles, LDS B panel) ----
    {
        dim3 blk(32, 4);
        dim3 grd(gridM, HID_DIM / 32);
        gnn_gemm_wmma<128><<<grd, blk, 0, stream>>>(x, W1, bufA, N, HID_DIM);
    }
    hipMemsetAsync(bufB, 0, (size_t)N * 128 * sizeof(float), stream);
    {
        dim3 blk(32, 8);  // 8 edges per block, one wave each
        gnn_scatter_edges<128><<<(E + 7) / 8, blk, 0, stream>>>(bufA, srcp, dstp, dinv, bufB, E);
    }
    gnn_fuse_bias<128, true><<<(N * 128 + 255) / 256, 256, 0, stream>>>(bufB, bufA, dinv, b1, bufC, N);

    // ---- layer 2: h2 = relu(h1') @ W2 ----
    {
        dim3 blk(32, 4);
        dim3 grd(gridM, OUT_DIM / 32);
        gnn_gemm_wmma<128><<<grd, blk, 0, stream>>>(bufC, W2, bufA, N, OUT_DIM);
    }
    hipMemsetAsync(bufB, 0, (size_t)N * 64 * sizeof(float), stream);
    {
        dim3 blk(32, 8);
        gnn_scatter_edges<64><<<(E + 7) / 8, blk, 0, stream>>>(bufA, srcp, dstp, dinv, bufB, E);
    }
    gnn_fuse_bias<64, false><<<(N * 64 + 255) / 256, 256, 0, stream>>>(bufB, bufA, dinv, b2, bufC, N);

    // ---- global mean pool ----
    hipMemsetAsync(out, 0, (size_t)N_GRAPHS * OUT_DIM * sizeof(float), stream);
    hipMemsetAsync(cnt, 0, (size_t)N_GRAPHS * sizeof(float), stream);
    gnn_pool_accum<<<N, OUT_DIM, 0, stream>>>(bufC, batch, out, cnt, N);
    gnn_pool_div<<<N_GRAPHS, OUT_DIM, 0, stream>>>(out, cnt);
}